// Entity_Knowledge_82179904242105
// MI455X (gfx1250) — compile-verified
//
#include <hip/hip_runtime.h>

#define VOCAB    20000
#define TOP_K    8
#define GLOVE    300
#define OUT_C    100
#define BB       32
#define LL       128
#define N_LABELS 8
#define NPOS     (BB * LL)        // 4096 positions
#define KPAD     320              // K padded (LDS row stride, zero-filled tail)
#define ROWLEN   (2 * GLOVE)      // 600 floats per (vocab,k) row
#define FACTSZ   (TOP_K * ROWLEN) // 4800 floats per vocab entry

typedef __attribute__((ext_vector_type(2)))  float    v2f;
typedef __attribute__((ext_vector_type(8)))  float    v8f;
typedef __attribute__((ext_vector_type(16))) _Float16 v16h;

// One block = one M-tile of 16 positions. blockDim = 320 (10 wave32 waves).
//   Phase A: threads 0..299 each own ONE channel c; for each of 16 positions,
//            accumulate fact_table[id][k][c] + [c+300] over 8 labels x 8 k
//            (perfectly balanced, fully coalesced 4B/lane loads), scale by
//            1/128, write to LDS. Threads 300..319 zero-fill the K pad.
//   Phase B: waves 0..6 each compute one 16x16 output N-tile with
//            V_WMMA_F32_16X16X4_F32; A fragments via ds_load from LDS,
//            B fragments from conv_w (120 KB, L2-resident).
__global__ void __launch_bounds__(320)
entity_knowledge_fused(const int*   __restrict__ labels, // [NPOS, N_LABELS]
                       const float* __restrict__ table,  // [VOCAB, TOP_K, 600]
                       const float* __restrict__ W,      // [OUT_C, GLOVE]
                       const float* __restrict__ bias,   // [OUT_C]
                       float*       __restrict__ Y)      // [NPOS, OUT_C]
{
    __shared__ float Xs[16][KPAD];   // 20 KB of 320 KB LDS

    const int mtile = blockIdx.x;    // 0..255
    const int tid   = threadIdx.x;   // 0..319

    // ---------------- Phase A: gather + mean -> LDS ----------------
    const float inv = 1.0f / 128.0f;     // mean over n_labels * 2 * top_k rows
    const int   c   = tid;               // owned channel (active when < 300)
    for (int p = 0; p < 16; ++p) {
        const int pos = mtile * 16 + p;
        if (c < GLOVE) {
            float accA = 0.0f, accB = 0.0f;   // split chain: even/odd k rows
            int id = labels[pos * N_LABELS + 0];
            #pragma unroll
            for (int j = 0; j < N_LABELS; ++j) {
                const float* base = table + (size_t)id * FACTSZ;
                if (j + 1 < N_LABELS) {
                    id = labels[pos * N_LABELS + j + 1];
                    // prefetch the next (random) vocab entry's first row pair
                    const float* nb = table + (size_t)id * FACTSZ;
                    __builtin_prefetch(nb + c, 0, 0);           // global_prefetch_b8
                    __builtin_prefetch(nb + c + GLOVE, 0, 0);
                }
                #pragma unroll
                for (int k = 0; k < TOP_K; k += 2) {
                    const float* r0 = base + (k + 0) * ROWLEN;
                    const float* r1 = base + (k + 1) * ROWLEN;
                    accA += r0[c] + r0[c + GLOVE];
                    accB += r1[c] + r1[c + GLOVE];
                }
            }
            Xs[p][c] = (accA + accB) * inv;
        } else {
            Xs[p][c] = 0.0f;                  // zero-pad channels 300..319
        }
    }
    __syncthreads();

    // ---------------- Phase B: WMMA GEMM out of LDS ----------------
    const int wave = tid >> 5;               // 0..9; waves 0..6 compute tiles
    const int lane = tid & 31;
    const int half = lane >> 4;              // hi/lo half-wave (K striping)
    const int l16  = lane & 15;

    if (wave < (OUT_C + 15) / 16) {          // 7 N-tiles cover 100 channels
        const int ntile = wave;
        const int col   = ntile * 16 + l16;            // output channel 0..111
        const int wcol  = (col < OUT_C) ? col : 0;     // clamp reads; cols >=100
                                                       // compute garbage that is
                                                       // simply never stored
        v8f cacc = {};
        for (int k = 0; k < GLOVE; k += 4) {
            // A 16x4 (M = l16): lanes 0-15 hold K={0,1}, lanes 16-31 K={2,3}
            v2f a;
            a.x = Xs[l16][k + half * 2 + 0];
            a.y = Xs[l16][k + half * 2 + 1];
            // B 4x16 (N = l16): B[k'][n] = W[n][k'], mirrored K striping
            const float* wr = W + (size_t)wcol * GLOVE + k + half * 2;
            v2f b;
            b.x = wr[0];
            b.y = wr[1];
            cacc = __builtin_amdgcn_wmma_f32_16x16x4_f32(false, a, false, b,
                                                         (short)0, cacc,
                                                         false, false);
        }
        // C/D layout: VGPR r -> row M = r + 8*half, col N = l16
        if (col < OUT_C) {
            const float bv = bias[col];
            #pragma unroll
            for (int r = 0; r < 8; ++r) {
                const int orow = mtile * 16 + r + half * 8;
                Y[(size_t)orow * OUT_C + col] = cacc[r] + bv;
            }
        }
    }
}

extern "C" void kernel_launch(void* const* d_in, const int* in_sizes, int n_in,
                              void* d_out, int out_size, void* d_ws, size_t ws_size,
                              hipStream_t stream) {
    (void)in_sizes; (void)n_in; (void)d_ws; (void)ws_size; (void)out_size;
    const int*   labels = (const int*)  d_in[0];   // detect_labels [32,128,8] i32
    const float* table  = (const float*)d_in[1];   // fact_table [20000,8,600] f32
    const float* W      = (const float*)d_in[2];   // conv_w [100,300] f32
    const float* bias   = (const float*)d_in[3];   // conv_b [100] f32
    float*       Y      = (float*)d_out;           // [32,128,100] f32

    entity_knowledge_fused<<<NPOS / 16, 320, 0, stream>>>(labels, table, W, bias, Y);
}